// SAGEConv_78340203479622
// MI455X (gfx1250) — compile-verified
//
#include <hip/hip_runtime.h>

#define NN 100000
#define NE 1600000
#define D  64
#define NT 8

typedef float v2f __attribute__((ext_vector_type(2)));
typedef float v8f __attribute__((ext_vector_type(8)));

// ---------------------------------------------------------------------------
// Kernel 1: zero workspace (neigh_sum [NN*64] + deg [NN]) = 6,500,000 floats
// ---------------------------------------------------------------------------
__global__ void sage_zero(float* __restrict__ ws, int n4) {
    int i = blockIdx.x * blockDim.x + threadIdx.x;
    if (i < n4) {
        ((float4*)ws)[i] = make_float4(0.f, 0.f, 0.f, 0.f);
    }
}

// ---------------------------------------------------------------------------
// Kernel 2: edge scatter. One wave per edge iteration: lane l handles columns
// 2l,2l+1 -> 256B coalesced gather of feat[src], hardware f32 atomics into
// neigh_sum[dst] (L2-resident). Lane 0 bumps the degree counter.
// ---------------------------------------------------------------------------
__global__ void sage_edges(const float* __restrict__ feat,
                           const int*   __restrict__ src,
                           const int*   __restrict__ dst,
                           float* __restrict__ nsum,
                           float* __restrict__ deg) {
    int wave   = (blockIdx.x * blockDim.x + threadIdx.x) >> 5;
    int lane   = threadIdx.x & 31;
    int nwaves = (gridDim.x * blockDim.x) >> 5;
    for (int e = wave; e < NE; e += nwaves) {
        int s = src[e];          // wave-uniform -> scalar load
        int d = dst[e];
        const float2 v = *(const float2*)&feat[s * D + lane * 2];
        float* p = &nsum[d * D + lane * 2];
        __hip_atomic_fetch_add(&p[0], v.x, __ATOMIC_RELAXED, __HIP_MEMORY_SCOPE_AGENT);
        __hip_atomic_fetch_add(&p[1], v.y, __ATOMIC_RELAXED, __HIP_MEMORY_SCOPE_AGENT);
        if (lane == 0) {
            __hip_atomic_fetch_add(&deg[d], 1.0f, __ATOMIC_RELAXED, __HIP_MEMORY_SCOPE_AGENT);
        }
    }
}

// ---------------------------------------------------------------------------
// Kernel 3: per 16-node tile (one wave each):
//   h_neigh = nsum * 1/max(deg,1)   (staged in LDS, pitch 68 -> bank-clean)
//   C = ntype@w_self^T + b_self + b_neigh   (K=8, plain FMAs into v8f acc)
//   C += h_neigh @ w_neigh^T via v_wmma_f32_16x16x4_f32  (16 K-steps x 4 N-tiles)
//   relu (per act_flag), store.
// blockDim=64 (2 waves), grid=3125 -> exactly 6250 tiles, no idle waves.
// ---------------------------------------------------------------------------
#define PITCH 68

__global__ __launch_bounds__(64) void sage_gemm(
        const float* __restrict__ ntype,
        const float* __restrict__ w_self,
        const float* __restrict__ b_self,
        const float* __restrict__ w_neigh,
        const float* __restrict__ b_neigh,
        const float* __restrict__ nsum,
        const float* __restrict__ deg,
        const int*   __restrict__ act_flag,
        float* __restrict__ out) {
    __shared__ __align__(16) float w_lds[D][PITCH];       // w_neigh row-major
    __shared__ __align__(16) float a_lds[2][16][PITCH];   // h_neigh tile per wave

    const int tid  = threadIdx.x;
    const int warp = tid >> 5;
    const int lane = tid & 31;
    const int hi   = lane >> 4;     // half-wave select (K pair / row+8)
    const int l15  = lane & 15;

    // Stage w_neigh (B matrix, column-major == row-major w_neigh) into LDS.
    for (int i = tid; i < D * D; i += 64) {
        w_lds[i >> 6][i & 63] = w_neigh[i];
    }

    const int tile = blockIdx.x * 2 + warp;
    const int m0   = tile * 16;

    // Stage h_neigh tile: row-by-row, fully coalesced float2 per lane.
    for (int r = 0; r < 16; ++r) {
        float inv = 1.0f / fmaxf(deg[m0 + r], 1.0f);   // uniform -> scalar load
        float2 v  = *(const float2*)&nsum[(m0 + r) * D + lane * 2];
        v.x *= inv;
        v.y *= inv;
        *(float2*)&a_lds[warp][r][lane * 2] = v;
    }
    __syncthreads();

    const int act = act_flag[0];

    // Hoist this lane's 8 ntype rows (rows m0+v+8*hi) into registers.
    float nt[8][NT];
#pragma unroll
    for (int v = 0; v < 8; ++v) {
        const int row = m0 + v + 8 * hi;
#pragma unroll
        for (int t = 0; t < NT; ++t) nt[v][t] = ntype[row * NT + t];
    }

    for (int n0 = 0; n0 < D; n0 += 16) {
        const int n = n0 + l15;

        float ws[NT];
#pragma unroll
        for (int t = 0; t < NT; ++t) ws[t] = w_self[n * NT + t];
        const float bias = b_self[n] + b_neigh[n];

        // Self-path directly in the WMMA C/D layout:
        // c[v] <-> (row m0+v+8*hi, col n0+l15)
        v8f c;
#pragma unroll
        for (int v = 0; v < 8; ++v) {
            float acc = bias;
#pragma unroll
            for (int t = 0; t < NT; ++t) acc += nt[v][t] * ws[t];
            c[v] = acc;
        }

        // Neighbor-path: 16 WMMA f32 16x16x4 steps over K=64.
#pragma unroll
        for (int k = 0; k < 16; ++k) {
            v2f a = *(const v2f*)&a_lds[warp][l15][4 * k + 2 * hi];
            v2f b = *(const v2f*)&w_lds[n][4 * k + 2 * hi];
            c = __builtin_amdgcn_wmma_f32_16x16x4_f32(
                    false, a, false, b, (short)0, c, false, false);
        }

#pragma unroll
        for (int v = 0; v < 8; ++v) {
            const int row = m0 + v + 8 * hi;
            float val = c[v];
            if (act) val = fmaxf(val, 0.0f);
            out[row * D + n] = val;
        }
    }
}

// ---------------------------------------------------------------------------
extern "C" void kernel_launch(void* const* d_in, const int* in_sizes, int n_in,
                              void* d_out, int out_size, void* d_ws, size_t ws_size,
                              hipStream_t stream) {
    const float* feat    = (const float*)d_in[0];
    const float* ntype   = (const float*)d_in[1];
    const float* w_self  = (const float*)d_in[2];
    const float* b_self  = (const float*)d_in[3];
    const float* w_neigh = (const float*)d_in[4];
    const float* b_neigh = (const float*)d_in[5];
    const int*   src     = (const int*)d_in[6];
    const int*   dst     = (const int*)d_in[7];
    const int*   act     = (const int*)d_in[8];
    float*       out     = (float*)d_out;

    float* nsum = (float*)d_ws;          // NN * D floats
    float* deg  = nsum + (size_t)NN * D; // NN floats
    // total scratch: (NN*D + NN)*4 = 26 MB

    // 1) zero accumulators
    const int n4 = (NN * D + NN) / 4;    // 1,625,000 float4s
    sage_zero<<<(n4 + 255) / 256, 256, 0, stream>>>(nsum, n4);

    // 2) edge gather + atomic scatter (L2-resident)
    sage_edges<<<2048, 256, 0, stream>>>(feat, src, dst, nsum, deg);

    // 3) mean + fused dual-GEMM + bias + relu, WMMA f32
    sage_gemm<<<NN / 32, 64, 0, stream>>>(ntype, w_self, b_self, w_neigh,
                                          b_neigh, nsum, deg, act, out);
}